// dual_mamba_19928648254254
// MI455X (gfx1250) — compile-verified
//
#include <hip/hip_runtime.h>
#include <math.h>

// ---- problem constants (match reference) ----
#define B_     4
#define C_     80
#define T_     800
#define DIN_   160
#define NST_   16
#define DTR_   5
#define KC_    4
#define ATT_   128
#define EMB_   192
#define DBW_   (DTR_ + 2*NST_)   // 37
#define BT_    (B_*T_)           // 3200
#define DEPTH_ 12

typedef __attribute__((ext_vector_type(2))) float v2f;
typedef __attribute__((ext_vector_type(8))) float v8f;

// ---------------- reductions ----------------
__device__ __forceinline__ float warp_sum(float v){
#pragma unroll
  for (int o = 16; o; o >>= 1) v += __shfl_xor(v, o, 32);
  return v;
}
__device__ __forceinline__ float warp_max(float v){
#pragma unroll
  for (int o = 16; o; o >>= 1) v = fmaxf(v, __shfl_xor(v, o, 32));
  return v;
}
__device__ float block_sum(float v, float* sm){
  int tid = threadIdx.x;
  v = warp_sum(v);
  if ((tid & 31) == 0) sm[tid >> 5] = v;
  __syncthreads();
  float r = 0.f;
  if (tid < (int)(blockDim.x >> 5)) r = sm[tid];
  if ((tid >> 5) == 0) r = warp_sum(r);
  if (tid == 0) sm[0] = r;
  __syncthreads();
  r = sm[0];
  __syncthreads();
  return r;
}
__device__ float block_max(float v, float* sm){
  int tid = threadIdx.x;
  v = warp_max(v);
  if ((tid & 31) == 0) sm[tid >> 5] = v;
  __syncthreads();
  float r = -3.4e38f;
  if (tid < (int)(blockDim.x >> 5)) r = sm[tid];
  if ((tid >> 5) == 0) r = warp_max(r);
  if (tid == 0) sm[0] = r;
  __syncthreads();
  r = sm[0];
  __syncthreads();
  return r;
}

// ---------------- WMMA f32 GEMM ----------------
// Out[M,N] = act(A[M,K] @ W[K,N] + bias) + residual ; M%16==0, K%4==0, N arbitrary.
// One wave computes one 16x16 tile via V_WMMA_F32_16X16X4_F32.
// A frag (16x4 f32): lanes 0-15 hold row M=lane, K={k,k+1}; lanes 16-31 hold row M=lane-16, K={k+2,k+3}.
// B frag (4x16 f32): lanes 0-15 v0/v1 = K={k,k+1}, N=lane; lanes 16-31 = K={k+2,k+3}, N=lane-16.
// C/D: VGPR r -> M = r + 8*(lane>=16), N = lane&15.
// Out-of-range columns (only when N%16!=0) are loaded CLAMPED to column N-1 instead of
// predicated: garbage there stays confined to its own output column, which is never stored.
__global__ __launch_bounds__(256)
void gemm_wmma_f32(const float* __restrict__ A, const float* __restrict__ W,
                   const float* __restrict__ bias, const float* __restrict__ residual,
                   float* __restrict__ Out, int M, int K, int N, int act)
{
  int lane = threadIdx.x & 31;
  int wave = blockIdx.x * (blockDim.x >> 5) + (threadIdx.x >> 5);
  int ntN = (N + 15) >> 4;
  int ntiles = (M >> 4) * ntN;
  if (wave >= ntiles) return;                 // wave-uniform exit
  int tm = wave / ntN, tn = wave % ntN;
  int m0 = tm << 4, n0 = tn << 4;
  int half = lane >> 4;
  int l15  = lane & 15;
  int nn   = n0 + l15;
  bool nok = (nn < N);
  int nc   = nok ? nn : (N - 1);              // clamped, always in-bounds

  const float* ap = A + (size_t)(m0 + l15) * K + 2*half;   // advances by 4 per step
  const float* wp = W + (size_t)(2*half) * N + nc;         // advances by 4*N per step

  v8f c = {0.f,0.f,0.f,0.f,0.f,0.f,0.f,0.f};
  int steps = K >> 2;
#pragma unroll 4
  for (int s = 0; s < steps; ++s){
    v2f a, b;
    a.x = ap[0];
    a.y = ap[1];
    b.x = wp[0];
    b.y = wp[N];
    c = __builtin_amdgcn_wmma_f32_16x16x4_f32(false, a, false, b, (short)0, c, false, false);
    ap += 4;
    wp += (size_t)4 * N;
  }

  if (nok){
#pragma unroll
    for (int r = 0; r < 8; ++r){
      int mm = m0 + r + 8*half;
      float v = c[r];
      if (bias)     v += bias[nn];
      if (act == 1) v = 0.5f * v * (1.f + erff(v * 0.70710678118654752f)); // exact GELU
      if (residual) v += residual[(size_t)mm * N + nn];
      Out[(size_t)mm * N + nn] = v;
    }
  }
}

static void launch_gemm(const float* A, const float* W, const float* bias, const float* residual,
                        float* Out, int M, int K, int N, int act, hipStream_t stream){
  int ntiles = (M/16) * ((N+15)/16);
  int blocks = (ntiles + 7) / 8;              // 8 waves / block
  gemm_wmma_f32<<<blocks, 256, 0, stream>>>(A, W, bias, residual, Out, M, K, N, act);
}

// ---------------- elementwise / stage kernels ----------------

// feat[B,C,T] -> x[B,T,C], subtract per-(b,c) mean over T (CMN)
__global__ void cmn_kernel(const float* __restrict__ feat, float* __restrict__ x){
  __shared__ float sm[8];
  int bc = blockIdx.x;                    // B_*C_
  int b = bc / C_, c = bc % C_;
  const float* row = feat + (size_t)bc * T_;
  float s = 0.f;
  for (int t = threadIdx.x; t < T_; t += blockDim.x) s += row[t];
  s = block_sum(s, sm);
  float mu = s * (1.f / T_);
  for (int t = threadIdx.x; t < T_; t += blockDim.x)
    x[(size_t)(b*T_ + t)*C_ + c] = row[t] - mu;
}

// LayerNorm over C_=80, one wave per row
__global__ void layernorm_kernel(const float* __restrict__ in, const float* __restrict__ g,
                                 const float* __restrict__ bt, float* __restrict__ out, int M){
  int lane = threadIdx.x & 31;
  int row  = blockIdx.x * (blockDim.x >> 5) + (threadIdx.x >> 5);
  if (row >= M) return;
  const float* r = in + (size_t)row * C_;
  float s = 0.f, s2 = 0.f;
  for (int c = lane; c < C_; c += 32){ float v = r[c]; s += v; s2 += v*v; }
  s = warp_sum(s); s2 = warp_sum(s2);
  float mu  = s * (1.f / C_);
  float var = s2 * (1.f / C_) - mu*mu;
  float inv = rsqrtf(var + 1e-5f);
  for (int c = lane; c < C_; c += 32)
    out[(size_t)row*C_ + c] = (r[c] - mu) * inv * g[c] + bt[c];
}

// time-flip [B,T,C]
__global__ void flip_kernel(const float* __restrict__ in, float* __restrict__ out){
  int id = blockIdx.x*blockDim.x + threadIdx.x;
  if (id >= BT_*C_) return;
  int c = id % C_, bt = id / C_;
  int b = bt / T_, t = bt % T_;
  out[(size_t)(b*T_ + (T_-1-t))*C_ + c] = in[id];
}

// depthwise causal conv (K=4) + SiLU ; xc = first half of xz rows
__global__ void conv_silu_kernel(const float* __restrict__ xz, const float* __restrict__ w,
                                 const float* __restrict__ bias, float* __restrict__ out){
  int id = blockIdx.x*blockDim.x + threadIdx.x;
  if (id >= BT_*DIN_) return;
  int d = id % DIN_, bt = id / DIN_;
  int b = bt / T_, t = bt % T_;
  float acc = bias[d];
#pragma unroll
  for (int k = 0; k < KC_; ++k){
    int ts = t - (KC_-1) + k;
    if (ts >= 0) acc += w[k*DIN_ + d] * xz[(size_t)(b*T_ + ts)*(2*DIN_) + d];
  }
  out[id] = acc / (1.f + __expf(-acc));
}

// dt = softplus(dbc[:, :5] @ dt_w + dt_b)
__global__ void dt_kernel(const float* __restrict__ dbc, const float* __restrict__ dtw,
                          const float* __restrict__ dtb, float* __restrict__ dt){
  int id = blockIdx.x*blockDim.x + threadIdx.x;
  if (id >= BT_*DIN_) return;
  int d = id % DIN_, bt = id / DIN_;
  float acc = dtb[d];
#pragma unroll
  for (int r = 0; r < DTR_; ++r) acc += dbc[(size_t)bt*DBW_ + r] * dtw[r*DIN_ + d];
  dt[id] = acc > 0.f ? acc + log1pf(__expf(-acc)) : log1pf(__expf(acc));
}

// selective scan: lane layout = 16 state-lanes (n) per (b,d) pair; y = sum_n h*C via shfl tree.
// Epilogue fuses +x*D and *silu(z).
__global__ void scan_kernel(const float* __restrict__ xcc, const float* __restrict__ dt,
                            const float* __restrict__ dbc, const float* __restrict__ xz,
                            const float* __restrict__ A_log, const float* __restrict__ Dp,
                            float* __restrict__ y){
  int id = blockIdx.x*blockDim.x + threadIdx.x;   // B_*DIN_*NST_ = 10240
  if (id >= B_*DIN_*NST_) return;
  int n    = id & 15;
  int pair = id >> 4;
  int d = pair % DIN_;
  int b = pair / DIN_;
  float An = -__expf(A_log[d*NST_ + n]);
  float Dd = Dp[d];
  float h = 0.f;
  for (int t = 0; t < T_; ++t){
    int bt = b*T_ + t;
    float xv  = xcc[(size_t)bt*DIN_ + d];
    float dtv = dt [(size_t)bt*DIN_ + d];
    const float* row = dbc + (size_t)bt*DBW_;
    h = __expf(dtv*An)*h + (dtv*xv) * row[DTR_ + n];
    float p = h * row[DTR_ + NST_ + n];
#pragma unroll
    for (int o = 1; o < 16; o <<= 1) p += __shfl_xor(p, o, 32);
    if (n == 0){
      float z  = xz[(size_t)bt*(2*DIN_) + DIN_ + d];
      float sz = z / (1.f + __expf(-z));
      y[(size_t)bt*DIN_ + d] = (p + xv*Dd) * sz;
    }
  }
}

// fwd + time-flipped bwd
__global__ void combine_kernel(const float* __restrict__ f, const float* __restrict__ bw,
                               float* __restrict__ out){
  int id = blockIdx.x*blockDim.x + threadIdx.x;
  if (id >= BT_*C_) return;
  int c = id % C_, bt = id / C_;
  int b = bt / T_, t = bt % T_;
  out[id] = f[id] + bw[(size_t)(b*T_ + (T_-1-t))*C_ + c];
}

// per-(b,c) mean/std over T of x[B,T,C]
__global__ void meanstd_kernel(const float* __restrict__ x, float* __restrict__ mean,
                               float* __restrict__ stdv){
  __shared__ float sm[8];
  int bc = blockIdx.x;
  int b = bc / C_, c = bc % C_;
  float s = 0.f, s2 = 0.f;
  for (int t = threadIdx.x; t < T_; t += blockDim.x){
    float v = x[(size_t)(b*T_+t)*C_ + c];
    s += v; s2 += v*v;
  }
  s = block_sum(s, sm); s2 = block_sum(s2, sm);
  if (threadIdx.x == 0){
    float mu  = s * (1.f / T_);
    float var = s2 * (1.f / T_) - mu*mu;
    mean[bc] = mu;
    stdv[bc] = sqrtf(fmaxf(var, 1e-12f));
  }
}

// g[bt, 0:80]=x, [80:160]=mean(b), [160:240]=std(b)
__global__ void buildg_kernel(const float* __restrict__ x, const float* __restrict__ mean,
                              const float* __restrict__ stdv, float* __restrict__ g){
  int id = blockIdx.x*blockDim.x + threadIdx.x;
  if (id >= BT_*3*C_) return;
  int c3 = id % (3*C_), bt = id / (3*C_);
  int b = bt / T_;
  float v;
  if (c3 < C_)        v = x[(size_t)bt*C_ + c3];
  else if (c3 < 2*C_) v = mean[b*C_ + (c3 - C_)];
  else                v = stdv[b*C_ + (c3 - 2*C_)];
  g[id] = v;
}

// relu -> BN1 -> tanh (in place on h1[bt, ATT_])
__global__ void attmid_kernel(float* __restrict__ h1, const float* __restrict__ g,
                              const float* __restrict__ bt, const float* __restrict__ m,
                              const float* __restrict__ vv){
  int id = blockIdx.x*blockDim.x + threadIdx.x;
  if (id >= BT_*ATT_) return;
  int d = id % ATT_;
  float v = fmaxf(h1[id], 0.f);
  v = (v - m[d]) * rsqrtf(vv[d] + 1e-5f) * g[d] + bt[d];
  h1[id] = tanhf(v);
}

// softmax over T per (b,c), weighted mean/std -> pooled[B, 2C]
__global__ void pool_kernel(const float* __restrict__ a, const float* __restrict__ x,
                            float* __restrict__ pooled){
  __shared__ float sm[8];
  int bc = blockIdx.x;
  int b = bc / C_, c = bc % C_;
  float mx = -3.4e38f;
  for (int t = threadIdx.x; t < T_; t += blockDim.x)
    mx = fmaxf(mx, a[(size_t)(b*T_+t)*C_ + c]);
  mx = block_max(mx, sm);
  float s = 0.f;
  for (int t = threadIdx.x; t < T_; t += blockDim.x)
    s += __expf(a[(size_t)(b*T_+t)*C_ + c] - mx);
  s = block_sum(s, sm);
  float inv = 1.f / s;
  float mu = 0.f, m2 = 0.f;
  for (int t = threadIdx.x; t < T_; t += blockDim.x){
    float w  = __expf(a[(size_t)(b*T_+t)*C_ + c] - mx) * inv;
    float xv = x[(size_t)(b*T_+t)*C_ + c];
    mu += w*xv; m2 += w*xv*xv;
  }
  mu = block_sum(mu, sm); m2 = block_sum(m2, sm);
  if (threadIdx.x == 0){
    pooled[b*(2*C_) + c]      = mu;
    pooled[b*(2*C_) + C_ + c] = sqrtf(fmaxf(m2 - mu*mu, 1e-12f));
  }
}

// BN2 (inline) + FC -> out[B, EMB]
__global__ void final_kernel(const float* __restrict__ pooled, const float* __restrict__ g,
                             const float* __restrict__ bt, const float* __restrict__ m,
                             const float* __restrict__ vv, const float* __restrict__ fw,
                             const float* __restrict__ fb, float* __restrict__ out){
  int id = blockIdx.x*blockDim.x + threadIdx.x;
  if (id >= B_*EMB_) return;
  int e = id % EMB_, b = id / EMB_;
  float acc = fb[e];
  for (int i = 0; i < 2*C_; ++i){
    float pv = pooled[b*2*C_ + i];
    pv = (pv - m[i]) * rsqrtf(vv[i] + 1e-5f) * g[i] + bt[i];
    acc += pv * fw[i*EMB_ + e];
  }
  out[id] = acc;
}

// ---------------- host orchestration ----------------
extern "C" void kernel_launch(void* const* d_in, const int* in_sizes, int n_in,
                              void* d_out, int out_size, void* d_ws, size_t ws_size,
                              hipStream_t stream){
  const float* feat    = (const float*)d_in[0];
  const float* ln1_g   = (const float*)d_in[1];
  const float* ln1_b   = (const float*)d_in[2];
  const float* in_proj = (const float*)d_in[3];
  const float* conv_w  = (const float*)d_in[4];
  const float* conv_b  = (const float*)d_in[5];
  const float* x_proj  = (const float*)d_in[6];
  const float* dt_w    = (const float*)d_in[7];
  const float* dt_b    = (const float*)d_in[8];
  const float* A_log   = (const float*)d_in[9];
  const float* Dpar    = (const float*)d_in[10];
  const float* out_proj= (const float*)d_in[11];
  const float* ln2_g   = (const float*)d_in[12];
  const float* ln2_b   = (const float*)d_in[13];
  const float* ff_w1   = (const float*)d_in[14];
  const float* ff_b1   = (const float*)d_in[15];
  const float* ff_w2   = (const float*)d_in[16];
  const float* ff_b2   = (const float*)d_in[17];
  const float* att_w1  = (const float*)d_in[18];
  const float* att_b1  = (const float*)d_in[19];
  const float* bn1_g   = (const float*)d_in[20];
  const float* bn1_b   = (const float*)d_in[21];
  const float* bn1_m   = (const float*)d_in[22];
  const float* bn1_v   = (const float*)d_in[23];
  const float* att_w2  = (const float*)d_in[24];
  const float* att_b2  = (const float*)d_in[25];
  const float* bn2_g   = (const float*)d_in[26];
  const float* bn2_b   = (const float*)d_in[27];
  const float* bn2_m   = (const float*)d_in[28];
  const float* bn2_v   = (const float*)d_in[29];
  const float* fc_w    = (const float*)d_in[30];
  const float* fc_b    = (const float*)d_in[31];
  float* out = (float*)d_out;

  // workspace carve-out (floats)
  float* p = (float*)d_ws;
  float* x    = p; p += BT_*C_;
  float* xn   = p; p += BT_*C_;
  float* xnf  = p; p += BT_*C_;
  float* xz   = p; p += BT_*2*DIN_;
  float* xcc  = p; p += BT_*DIN_;
  float* dbc  = p; p += BT_*DBW_;
  float* dtv  = p; p += BT_*DIN_;
  float* yv   = p; p += BT_*DIN_;
  float* mo_f = p; p += BT_*C_;
  float* mo_b = p; p += BT_*C_;
  float* sbuf = p; p += BT_*C_;
  float* mbuf = p; p += BT_*C_;
  float* ff1  = p; p += BT_*4*C_;
  float* gbuf = p; p += BT_*3*C_;
  float* h1   = p; p += BT_*ATT_;
  float* abuf = p; p += BT_*C_;
  float* pooled = p; p += B_*2*C_;
  float* meanb  = p; p += B_*C_;
  float* stdb   = p; p += B_*C_;

  const int EW = 256;
  cmn_kernel<<<B_*C_, EW, 0, stream>>>(feat, x);

  for (int l = 0; l < DEPTH_; ++l){
    const float* Lln1g = ln1_g   + (size_t)l*C_;
    const float* Lln1b = ln1_b   + (size_t)l*C_;
    const float* Lip   = in_proj + (size_t)l*C_*2*DIN_;
    const float* Lcw   = conv_w  + (size_t)l*KC_*DIN_;
    const float* Lcb   = conv_b  + (size_t)l*DIN_;
    const float* Lxp   = x_proj  + (size_t)l*DIN_*DBW_;
    const float* Ldtw  = dt_w    + (size_t)l*DTR_*DIN_;
    const float* Ldtb  = dt_b    + (size_t)l*DIN_;
    const float* LA    = A_log   + (size_t)l*DIN_*NST_;
    const float* LD    = Dpar    + (size_t)l*DIN_;
    const float* Lop   = out_proj+ (size_t)l*DIN_*C_;
    const float* Lln2g = ln2_g   + (size_t)l*C_;
    const float* Lln2b = ln2_b   + (size_t)l*C_;
    const float* Lf1w  = ff_w1   + (size_t)l*C_*4*C_;
    const float* Lf1b  = ff_b1   + (size_t)l*4*C_;
    const float* Lf2w  = ff_w2   + (size_t)l*4*C_*C_;
    const float* Lf2b  = ff_b2   + (size_t)l*C_;

    layernorm_kernel<<<BT_/8, EW, 0, stream>>>(x, Lln1g, Lln1b, xn, BT_);
    flip_kernel<<<(BT_*C_+EW-1)/EW, EW, 0, stream>>>(xn, xnf);

    for (int dir = 0; dir < 2; ++dir){
      const float* src = dir ? xnf : xn;
      float*       mo  = dir ? mo_b : mo_f;
      launch_gemm(src, Lip, nullptr, nullptr, xz, BT_, C_, 2*DIN_, 0, stream);
      conv_silu_kernel<<<(BT_*DIN_+EW-1)/EW, EW, 0, stream>>>(xz, Lcw, Lcb, xcc);
      launch_gemm(xcc, Lxp, nullptr, nullptr, dbc, BT_, DIN_, DBW_, 0, stream);
      dt_kernel<<<(BT_*DIN_+EW-1)/EW, EW, 0, stream>>>(dbc, Ldtw, Ldtb, dtv);
      scan_kernel<<<(B_*DIN_*NST_+EW-1)/EW, EW, 0, stream>>>(xcc, dtv, dbc, xz, LA, LD, yv);
      launch_gemm(yv, Lop, nullptr, nullptr, mo, BT_, DIN_, C_, 0, stream);
    }

    combine_kernel<<<(BT_*C_+EW-1)/EW, EW, 0, stream>>>(mo_f, mo_b, sbuf);
    layernorm_kernel<<<BT_/8, EW, 0, stream>>>(sbuf, Lln2g, Lln2b, mbuf, BT_);
    launch_gemm(mbuf, Lf1w, Lf1b, nullptr, ff1, BT_, C_, 4*C_, 1, stream);      // GELU
    launch_gemm(ff1,  Lf2w, Lf2b, x,      x,   BT_, 4*C_, C_, 0, stream);       // + residual (in place)
  }

  meanstd_kernel<<<B_*C_, EW, 0, stream>>>(x, meanb, stdb);
  buildg_kernel<<<(BT_*3*C_+EW-1)/EW, EW, 0, stream>>>(x, meanb, stdb, gbuf);
  launch_gemm(gbuf, att_w1, att_b1, nullptr, h1,   BT_, 3*C_, ATT_, 0, stream);
  attmid_kernel<<<(BT_*ATT_+EW-1)/EW, EW, 0, stream>>>(h1, bn1_g, bn1_b, bn1_m, bn1_v);
  launch_gemm(h1,   att_w2, att_b2, nullptr, abuf, BT_, ATT_, C_,  0, stream);
  pool_kernel<<<B_*C_, EW, 0, stream>>>(abuf, x, pooled);
  final_kernel<<<(B_*EMB_+EW-1)/EW, EW, 0, stream>>>(pooled, bn2_g, bn2_b, bn2_m, bn2_v,
                                                     fc_w, fc_b, out);
}